// STDP_33260226740731
// MI455X (gfx1250) — compile-verified
//
#include <hip/hip_runtime.h>

// ---------------- problem constants (match reference) ----------------
#define TT        32
#define C_IN      64
#define HH        128
#define WW        128
#define C_OUT     128
#define KH        5
#define KW        5
#define H_OUT     124
#define W_OUT     124
#define K_WIN     256
#define PATCH     (C_IN * KH * KW)      // 1600
#define LR_PLUS   0.004f
#define LR_MINUS  -0.003f

#define IN_T_STRIDE   (C_IN * HH * WW)          // 1,048,576
#define OUT_T_STRIDE  (C_OUT * H_OUT * W_OUT)   // 1,968,128

typedef __attribute__((ext_vector_type(2))) float v2f;
typedef __attribute__((ext_vector_type(8))) float v8f;

// One block per output channel f.
// Phase 1 (thread 0): find LAST winner with winners[k].f == f (scan overwrite
//   semantics of lax.scan), and T-reduce out_lat at that (f,r,c).
// Phase 2 (8 waves): T-reduce the 64x5x5 input patch with accumulating
//   V_WMMA_F32_16X16X4_F32 (A = ones 16x4, B = 4 timesteps x 16 positions),
//   16 positions per wave per pass, 8 WMMAs cover T=32 exactly.
// Phase 3: lr = (patch_sum >= out_val) ? LR+ : LR-;  w' = clip(w + lr*w*(1-w)).
__global__ __launch_bounds__(256) void stdp_update_kernel(
    const float* __restrict__ in_spk,   // (T, C_IN, H, W)
    const float* __restrict__ out_spk,  // (T, C_OUT, H_OUT, W_OUT)
    const float* __restrict__ weight,   // (C_OUT, C_IN, KH, KW)
    const int*   __restrict__ winners,  // (K_WIN, 3) = [f, r, c]
    float*       __restrict__ out)      // (C_OUT, C_IN, KH, KW)
{
    const int f   = blockIdx.x;
    const int tid = threadIdx.x;

    __shared__ int   s_win, s_r, s_c;
    __shared__ float s_outval;

    if (tid == 0) {
        int win = -1;
        for (int k = 0; k < K_WIN; ++k)
            if (winners[3 * k] == f) win = k;         // last occurrence wins
        s_win = win;
        if (win >= 0) {
            int r = winners[3 * win + 1];
            int c = winners[3 * win + 2];
            s_r = r; s_c = c;
            float v = 0.0f;                            // out_lat[f, r, c]
            const float* p = out_spk + (size_t)f * (H_OUT * W_OUT)
                                     + (size_t)r * W_OUT + c;
            #pragma unroll
            for (int t = 0; t < TT; ++t)
                v += p[(size_t)t * OUT_T_STRIDE];
            s_outval = v;
        }
    }
    __syncthreads();

    const int win = s_win;
    if (win < 0) {                                     // lr == 0 for this channel
        for (int p = tid; p < PATCH; p += 256) {
            float w = weight[(size_t)f * PATCH + p];
            out[(size_t)f * PATCH + p] = fminf(fmaxf(w, 0.0f), 1.0f);
        }
        return;
    }

    const int   r      = s_r;
    const int   c      = s_c;
    const float outval = s_outval;

    const int wave  = tid >> 5;        // 0..7 (wave32)
    const int lane  = tid & 31;
    const int n     = lane & 15;       // column 0..15 within the WMMA tile
    const int thalf = (lane >> 4) * 2; // lanes 16-31 cover timesteps +2,+3

    v2f ones; ones.x = 1.0f; ones.y = 1.0f;            // A = all-ones 16x4

    // 13 passes x (8 waves x 16 positions) = 1664 >= 1600, tail masked.
    for (int pass = 0; pass < 13; ++pass) {
        const int  p     = pass * 128 + wave * 16 + n;
        const bool valid = (p < PATCH);
        const int  pc    = valid ? p : (PATCH - 1);    // clamp address, mask store

        const int cin = pc / (KH * KW);
        const int rem = pc - cin * (KH * KW);
        const int kh  = rem / KW;
        const int kw  = rem - kh * KW;
        const float* base = in_spk + (size_t)cin * (HH * WW)
                                   + (size_t)(r + kh) * WW + (c + kw);

        v8f acc = {};
        #pragma unroll
        for (int chunk = 0; chunk < 8; ++chunk) {      // 8 * K4 = T32
            const int t0 = chunk * 4 + thalf;
            v2f b;
            b.x = base[(size_t)t0       * IN_T_STRIDE];
            b.y = base[(size_t)(t0 + 1) * IN_T_STRIDE];
            acc = __builtin_amdgcn_wmma_f32_16x16x4_f32(
                /*neg_a=*/false, ones, /*neg_b=*/false, b,
                /*c_mod=*/(short)0, acc, /*reuse_a=*/false, /*reuse_b=*/false);
        }

        // Every D row equals the column sums; lane<16 reads (M=0, N=n).
        if ((lane < 16) && valid) {
            const float ps = acc[0];                   // patch sum (exact int <= 32)
            const float lr = (ps >= outval) ? LR_PLUS : LR_MINUS;
            const float w  = weight[(size_t)f * PATCH + p];
            const float nw = w + lr * (w * (1.0f - w));
            out[(size_t)f * PATCH + p] = fminf(fmaxf(nw, 0.0f), 1.0f);
        }
    }
}

extern "C" void kernel_launch(void* const* d_in, const int* in_sizes, int n_in,
                              void* d_out, int out_size, void* d_ws, size_t ws_size,
                              hipStream_t stream) {
    (void)in_sizes; (void)n_in; (void)d_ws; (void)ws_size; (void)out_size;
    const float* in_spk  = (const float*)d_in[0];
    const float* out_spk = (const float*)d_in[1];
    const float* weight  = (const float*)d_in[2];
    const int*   winners = (const int*)d_in[3];
    float*       out     = (float*)d_out;

    stdp_update_kernel<<<dim3(C_OUT), dim3(256), 0, stream>>>(
        in_spk, out_spk, weight, winners, out);
}